// MultiScaleRoIAlign_68393059222177
// MI455X (gfx1250) — compile-verified
//
#include <hip/hip_runtime.h>
#include <math.h>

// MultiScaleRoIAlign for MI455X (gfx1250).
// One workgroup (256 threads = 8 wave32) per RoI. Per-RoI footprint of the
// selected FPN level is staged into LDS with gfx1250 async gather-to-LDS
// (ASYNCcnt), double-buffered across channel chunks (1 wave = 1 channel).

#define NOUT   7
#define NS     14              // NOUT * SR sample positions per axis
#define NCH    256
#define CHUNK  8               // channels staged per LDS buffer (== waves per WG)
#define NCHUNK (NCH / CHUNK)
#define FPMAX  1216            // max footprint floats per channel (bound ~900)

#if __has_builtin(__builtin_amdgcn_global_load_async_to_lds_b32)
// Probe-confirmed (round 1 diagnostic): param0 = addrspace(1) int* (global
// source), param1 = addrspace(3) int* (LDS destination), then imm offset, imm cpol.
#define ASYNC_B32(ldsdst, gsrc)                                                \
  __builtin_amdgcn_global_load_async_to_lds_b32(                               \
      (__attribute__((address_space(1))) int*)(gsrc),                          \
      (__attribute__((address_space(3))) int*)(ldsdst), 0, 0)
#else
#warning "no __builtin_amdgcn_global_load_async_to_lds_b32: synchronous staging fallback"
#define ASYNC_B32(ldsdst, gsrc) (*(ldsdst) = *(gsrc))
#endif

#if __has_builtin(__builtin_amdgcn_s_wait_asynccnt)
#define WAIT_ASYNC0() __builtin_amdgcn_s_wait_asynccnt(0)
#else
#warning "no __builtin_amdgcn_s_wait_asynccnt: inline-asm fallback"
#define WAIT_ASYNC0() asm volatile("s_wait_asynccnt 0" ::: "memory")
#endif

// Legacy (aligned=False) RoIAlign sample along one axis, matching the
// reference's clamp + validity semantics. Invalid samples get zero weights.
__device__ __forceinline__ void sample_axis(float o, float binsz, int L, int i,
                                            int& lo, int& hi,
                                            float& wlo, float& whi) {
  int p = i >> 1;
  float off = (i & 1) ? 0.75f : 0.25f;
  float y = o + ((float)p + off) * binsz;
  bool valid = (y >= -1.0f) && (y <= (float)L);
  y = fmaxf(y, 0.0f);
  int yl = (int)y;                 // y >= 0 -> trunc == floor
  int yh;
  if (yl >= L - 1) { yl = L - 1; yh = L - 1; y = (float)yl; }
  else             { yh = yl + 1; }
  float ly = y - (float)yl;
  float hy = 1.0f - ly;
  wlo = valid ? hy : 0.0f;
  whi = valid ? ly : 0.0f;
  lo = yl; hi = yh;
}

__global__ __launch_bounds__(256, 1)
void msroi_align_kernel(const float* __restrict__ f0, const float* __restrict__ f1,
                        const float* __restrict__ f2, const float* __restrict__ f3,
                        const float* __restrict__ boxes, float* __restrict__ out,
                        int nPerB) {
  __shared__ float sWy0[16], sWy1[16], sWx0[16], sWx1[16];
  __shared__ int   sYl[16], sYh[16], sXl[16], sXh[16];
  __shared__ float sBuf[2][CHUNK * FPMAX];   // double-buffered footprint, ~76 KB

  const int r = blockIdx.x;
  const int b = r / nPerB;
  const float bx1 = boxes[r * 4 + 0], by1 = boxes[r * 4 + 1];
  const float bx2 = boxes[r * 4 + 2], by2 = boxes[r * 4 + 3];

  // FPN level mapping: clip(floor(4 + log2(sqrt(area)/224 + 1e-6)), 2, 5) - 2
  float area = (bx2 - bx1) * (by2 - by1);
  float sz = sqrtf(fmaxf(area, 0.0f));
  int lvl = (int)floorf(4.0f + log2f(sz / 224.0f + 1e-6f));
  lvl = lvl < 2 ? 2 : (lvl > 5 ? 5 : lvl);
  lvl -= 2;

  const int   sizesArr[4]  = {200, 100, 50, 25};
  const float scalesArr[4] = {0.25f, 0.125f, 0.0625f, 0.03125f};
  const int   L  = sizesArr[lvl];
  const float sc = scalesArr[lvl];
  const float* fsel = (lvl == 0) ? f0 : (lvl == 1) ? f1 : (lvl == 2) ? f2 : f3;
  const float* feat = fsel + (size_t)b * NCH * L * L;

  const float x1 = bx1 * sc, y1 = by1 * sc;
  const float roi_w = fmaxf(bx2 * sc - x1, 1.0f);
  const float roi_h = fmaxf(by2 * sc - y1, 1.0f);
  const float bw = roi_w / NOUT, bh = roi_h / NOUT;

  // Footprint bounds (sample coords are monotone in i).
  int lo0, hi0, loN, hiN; float du, dv;
  sample_axis(y1, bh, L, 0,      lo0, hi0, du, dv);
  sample_axis(y1, bh, L, NS - 1, loN, hiN, du, dv);
  const int y0 = lo0, fh = hiN - lo0 + 1;
  sample_axis(x1, bw, L, 0,      lo0, hi0, du, dv);
  sample_axis(x1, bw, L, NS - 1, loN, hiN, du, dv);
  const int x0 = lo0, fw = hiN - lo0 + 1;
  const int fpc = fh * fw;

  const int t = threadIdx.x;
  if (t < NS) {
    int lo, hi; float wl, wh;
    sample_axis(y1, bh, L, t, lo, hi, wl, wh);
    sYl[t] = lo; sYh[t] = hi; sWy0[t] = wl; sWy1[t] = wh;
  } else if (t >= 32 && t < 32 + NS) {
    int i = t - 32; int lo, hi; float wl, wh;
    sample_axis(x1, bw, L, i, lo, hi, wl, wh);
    sXl[i] = lo; sXh[i] = hi; sWx0[i] = wl; sWx1[i] = wh;
  }
  __syncthreads();

  const size_t HW = (size_t)L * L;

  if (fpc <= FPMAX) {            // staged path (uniform per block; always taken)
    const int cw   = t >> 5;     // wave id == channel-within-chunk
    const int lane = t & 31;

    // Prologue: async-stage chunk 0.
    {
      const float* g = feat + (size_t)cw * HW + (size_t)y0 * L + x0;
      float* lb = &sBuf[0][cw * fpc];
      for (int ry = 0; ry < fh; ++ry) {
        const float* grow = g + (size_t)ry * L;
        float* lrow = lb + ry * fw;
        for (int rx = lane; rx < fw; rx += 32)
          ASYNC_B32(lrow + rx, grow + rx);
      }
    }

    for (int k = 0; k < NCHUNK; ++k) {
      WAIT_ASYNC0();             // my chunk-k loads done (k+1 not yet issued)
      __syncthreads();           // everyone's chunk-k loads done; buf[(k+1)&1] free

      if (k + 1 < NCHUNK) {      // prefetch chunk k+1, overlapped with compute k
        const int c0 = (k + 1) * CHUNK;
        const float* g = feat + (size_t)(c0 + cw) * HW + (size_t)y0 * L + x0;
        float* lb = &sBuf[(k + 1) & 1][cw * fpc];
        for (int ry = 0; ry < fh; ++ry) {
          const float* grow = g + (size_t)ry * L;
          float* lrow = lb + ry * fw;
          for (int rx = lane; rx < fw; rx += 32)
            ASYNC_B32(lrow + rx, grow + rx);
        }
      }

      // Compute chunk k from LDS: wave cw handles channel k*CHUNK + cw.
      const float* ch = &sBuf[k & 1][cw * fpc];
      float* obase = out + ((size_t)r * NCH + (size_t)k * CHUNK + cw) * (NOUT * NOUT);
      for (int bin = lane; bin < NOUT * NOUT; bin += 32) {
        int ph = bin / NOUT, pw = bin - ph * NOUT;
        float acc = 0.0f;
#pragma unroll
        for (int s = 0; s < 2; ++s) {
          int sy = 2 * ph + s;
          int ry0 = sYl[sy] - y0, ry1 = sYh[sy] - y0;
          float wyA = sWy0[sy], wyB = sWy1[sy];
#pragma unroll
          for (int u = 0; u < 2; ++u) {
            int sx = 2 * pw + u;
            int cx0 = sXl[sx] - x0, cx1 = sXh[sx] - x0;
            float wxA = sWx0[sx], wxB = sWx1[sx];
            float vtop = wxA * ch[ry0 * fw + cx0] + wxB * ch[ry0 * fw + cx1];
            float vbot = wxA * ch[ry1 * fw + cx0] + wxB * ch[ry1 * fw + cx1];
            acc += wyA * vtop + wyB * vbot;
          }
        }
        obase[bin] = 0.25f * acc;
      }
    }
  } else {
    // Fallback: direct global gathers (footprint bound proof says unreachable).
    for (int idx = t; idx < NCH * NOUT * NOUT; idx += 256) {
      int c   = idx / (NOUT * NOUT);
      int bin = idx - c * (NOUT * NOUT);
      int ph = bin / NOUT, pw = bin - ph * NOUT;
      const float* ch = feat + (size_t)c * HW;
      float acc = 0.0f;
      for (int s = 0; s < 2; ++s) {
        int sy = 2 * ph + s;
        int ry0 = sYl[sy], ry1 = sYh[sy];
        float wyA = sWy0[sy], wyB = sWy1[sy];
        for (int u = 0; u < 2; ++u) {
          int sx = 2 * pw + u;
          int cx0 = sXl[sx], cx1 = sXh[sx];
          float wxA = sWx0[sx], wxB = sWx1[sx];
          float vtop = wxA * ch[(size_t)ry0 * L + cx0] + wxB * ch[(size_t)ry0 * L + cx1];
          float vbot = wxA * ch[(size_t)ry1 * L + cx0] + wxB * ch[(size_t)ry1 * L + cx1];
          acc += wyA * vtop + wyB * vbot;
        }
      }
      out[(size_t)r * NCH * (NOUT * NOUT) + idx] = 0.25f * acc;
    }
  }
}

extern "C" void kernel_launch(void* const* d_in, const int* in_sizes, int n_in,
                              void* d_out, int out_size, void* d_ws, size_t ws_size,
                              hipStream_t stream) {
  (void)n_in; (void)out_size; (void)d_ws; (void)ws_size;
  const float* f0    = (const float*)d_in[0];
  const float* f1    = (const float*)d_in[1];
  const float* f2    = (const float*)d_in[2];
  const float* f3    = (const float*)d_in[3];
  const float* boxes = (const float*)d_in[4];
  float* out = (float*)d_out;

  const int rois  = in_sizes[4] / 4;                 // 512
  const int B     = in_sizes[0] / (256 * 200 * 200); // 2
  const int nPerB = rois / (B > 0 ? B : 1);          // 256

  msroi_align_kernel<<<dim3(rois), dim3(256), 0, stream>>>(
      f0, f1, f2, f3, boxes, out, nPerB);
}